// NeedlemanWunschDecoder_36103495090637
// MI455X (gfx1250) — compile-verified
//
#include <hip/hip_runtime.h>
#include <stdint.h>

// ---------------------------------------------------------------------------
// Soft Needleman-Wunsch forward, B=128 independent 512x512 DPs.
// One workgroup (512 threads = 16 wave32) per batch element.
// 32x32 tiles, tile-grid wavefront across waves, in-register lane wavefront
// inside each tile.  theta/A tiles DMA'd into LDS with TENSOR_LOAD_TO_LDS
// (TDM), double buffered, synchronized with s_wait_tensorcnt.
// ---------------------------------------------------------------------------

#define NDIM      512
#define TILE      32
#define NTILE     (NDIM / TILE)   // 16
#define NWAVE     16
#define ROWSTRIDE 34              // interface row stride (33 used + 1 pad)

typedef unsigned int u32x4 __attribute__((ext_vector_type(4)));
typedef int          i32x8 __attribute__((ext_vector_type(8)));
typedef int          i32x4 __attribute__((ext_vector_type(4)));

__device__ __forceinline__ float fast_exp(float x) {
  return __builtin_amdgcn_exp2f(x * 1.44269504088896340736f);   // v_exp_f32
}
__device__ __forceinline__ float fast_log(float x) {
  return __builtin_amdgcn_logf(x) * 0.69314718055994530942f;    // v_log_f32
}

// Issue one TDM 2D tile load: 32x32 f32 tile from a row-major [512 x 512]
// tensor slice into LDS at byte offset lds_byte_off.  Descriptor per CDNA5
// ISA 8.3/8.4 (group0: count/lds_addr/global_addr/type; group1: data_size,
// tensor dims, tile dims, dim0 stride).  Groups 2/3 zero => 2D tensor.
// This toolchain's builtin takes 6 args (clang-23 form):
//   (u32x4 g0, i32x8 g1, i32x4 g2, i32x4 g3, i32x8 pad, i32 cpol)
__device__ __forceinline__ void tdm_load_tile_f32(const float* gsrc,
                                                  unsigned lds_byte_off) {
  unsigned long long ga = (unsigned long long)(uintptr_t)gsrc;
  unsigned galo = (unsigned)__builtin_amdgcn_readfirstlane((int)(ga & 0xFFFFFFFFull));
  unsigned gahi = (unsigned)__builtin_amdgcn_readfirstlane((int)(ga >> 32));
  unsigned loff = (unsigned)__builtin_amdgcn_readfirstlane((int)lds_byte_off);

  u32x4 g0;
  g0[0] = 1u;                                   // count=1 (valid user D#)
  g0[1] = loff;                                 // lds_addr (bytes)
  g0[2] = galo;                                 // global_addr[31:0]
  g0[3] = (gahi & 0x01FFFFFFu) | (2u << 30);    // global_addr[56:32] | type=2

  i32x8 g1;
  g1[0] = 0x00020000;                           // data_size=2 -> 4 bytes
  g1[1] = (int)(NDIM << 16);                    // tensor_dim0[15:0] @ [63:48]
  g1[2] = (int)(NDIM << 16);                    // tensor_dim1[15:0] @ [95:80]
  g1[3] = (int)(TILE << 16);                    // tile_dim0 @ [127:112]
  g1[4] = TILE;                                 // tile_dim1 @ [143:128]
  g1[5] = NDIM;                                 // tensor_dim0_stride[31:0]
  g1[6] = 0;
  g1[7] = 0;

  i32x4 gz4 = {0, 0, 0, 0};
  i32x8 gz8 = {0, 0, 0, 0, 0, 0, 0, 0};
  __builtin_amdgcn_tensor_load_to_lds(g0, g1, gz4, gz4, gz8, 0);
}

__global__ __launch_bounds__(NWAVE * 32, 1)
void nw_wavefront_kernel(const float* __restrict__ theta,
                         const float* __restrict__ A,
                         float* __restrict__ out) {
  extern __shared__ float smem[];
  const int b   = blockIdx.x;
  const int tid = threadIdx.x;
  const int w   = tid >> 5;    // wave id == tile row p
  const int c   = tid & 31;    // lane  == local column

  float* Hbuf = smem;                         // [NTILE][ROWSTRIDE] horiz. interfaces
  float* Vbuf = smem + NTILE * ROWSTRIDE;     // [NTILE][ROWSTRIDE] vert. interfaces
  float* tiles = smem + 2 * NTILE * ROWSTRIDE;

  const unsigned smem_base = (unsigned)(uintptr_t)smem;  // LDS byte offset of smem[0]
  const unsigned tiles_off = smem_base + 2u * NTILE * ROWSTRIDE * 4u;

  // Zero DP boundary interfaces: V[0,*] = V[*,0] = 0.
  for (int idx = tid; idx < 2 * NTILE * ROWSTRIDE; idx += blockDim.x)
    smem[idx] = 0.0f;
  __syncthreads();

  const float* thB = theta + (size_t)b * NDIM * NDIM;
  const float* aB  = A     + (size_t)b * NDIM * NDIM;

  const int p = w;
  float* waveTiles = tiles + (size_t)w * (2 * 2 * TILE * TILE);  // 2 bufs x (th,a)
  const unsigned waveTilesOff = tiles_off + (unsigned)(w * 2 * 2 * TILE * TILE) * 4u;
  float* VbufRow = Vbuf + p * ROWSTRIDE;      // private to wave p

  // Prologue: DMA tile (p, q=0) into buffer parity (w+0)&1.
  {
    const float* gth = thB + (size_t)(p * TILE) * NDIM;
    const float* gA  = aB  + (size_t)(p * TILE) * NDIM;
    unsigned o = waveTilesOff + (unsigned)((w & 1) * 2 * TILE * TILE) * 4u;
    tdm_load_tile_f32(gth, o);
    tdm_load_tile_f32(gA,  o + TILE * TILE * 4u);
  }

  for (int s = 0; s < 2 * NTILE - 1; ++s) {   // 31 tile-diagonal steps
    const int q  = s - w;
    const int qn = q + 1;
    const bool has_next = (qn >= 1) && (qn < NTILE);

    // Prefetch next tile on this wave's row (double buffered).
    if (has_next) {
      const float* gth = thB + (size_t)(p * TILE) * NDIM + qn * TILE;
      const float* gA  = aB  + (size_t)(p * TILE) * NDIM + qn * TILE;
      unsigned o = waveTilesOff + (unsigned)(((w + qn) & 1) * 2 * TILE * TILE) * 4u;
      tdm_load_tile_f32(gth, o);
      tdm_load_tile_f32(gA,  o + TILE * TILE * 4u);
    }

    if (q >= 0 && q < NTILE) {                // wave-uniform predicate
      // TDM ops are in-order per wave: leave the 2 prefetch loads in flight,
      // make sure the current tile's 2 loads have landed in LDS.
      if (has_next) __builtin_amdgcn_s_wait_tensorcnt(2);
      else          __builtin_amdgcn_s_wait_tensorcnt(0);

      const float* thT = waveTiles + ((w + q) & 1) * (2 * TILE * TILE);
      const float* aT  = thT + TILE * TILE;
      float* HbufRow = Hbuf + q * ROWSTRIDE;

      // Interfaces: top row (from tile above) + left col (from tile left).
      float topv      = HbufRow[1 + c];       // V[i0-1, j0+c]
      float cornerTop = HbufRow[0];           // V[i0-1, j0-1]
      float tshv = __shfl_up(topv, 1);        // V[i0-1, j0+c-1]
      if (c == 0) tshv = cornerTop;
      float wlast = VbufRow[TILE];            // V[i0+31, j0-1] (corner for tile below)

      float v1 = 0.0f, v2 = 0.0f;             // my V at substep kk-1 / kk-2
      for (int kk = 0; kk < 2 * TILE - 1; ++kk) {
        const int  r      = kk - c;           // local row this lane computes
        const bool active = (r >= 0) && (r < TILE);

        float l1 = __shfl_up(v1, 1);          // V[r,   c-1]
        float l2 = __shfl_up(v2, 1);          // V[r-1, c-1]

        int wi = kk + 1; if (wi > TILE) wi = TILE;   // clamp (unused when inactive)
        float wleft = VbufRow[wi];            // V[i0+kk, j0-1]   (lane 0)
        float wul   = VbufRow[wi - 1];        // V[i0+kk-1, j0-1] (lane 0)

        int rc = r < 0 ? 0 : (r > TILE - 1 ? TILE - 1 : r);
        float th = thT[rc * TILE + c];        // conflict-free anti-diagonal LDS read
        float av = aT [rc * TILE + c];

        float left = (c == 0) ? wleft : l1;
        float up, uplf;
        if (r == 0) { up = topv; uplf = tshv; }
        else        { up = v1;   uplf = (c == 0) ? wul : l2; }

        float x0 = av + up;                   // gap x
        float x1 = uplf;                      // match
        float x2 = av + left;                 // gap y
        float m  = fmaxf(fmaxf(x0, x1), x2);
        float vv = th + m + fast_log(fast_exp(x0 - m) + fast_exp(x1 - m) +
                                     fast_exp(x2 - m));
        if (active) {
          v2 = v1;
          v1 = vv;
          if (c == TILE - 1) VbufRow[1 + r] = vv;   // export right column
        }
      }
      // Export bottom row + corner for tile (p+1, q).
      HbufRow[1 + c] = v1;
      if (c == TILE - 1) HbufRow[0] = wlast;
    }
    __syncthreads();
  }

  if (tid == 0)
    out[b] = Hbuf[(NTILE - 1) * ROWSTRIDE + TILE];   // V[512, 512]
}

extern "C" void kernel_launch(void* const* d_in, const int* in_sizes, int n_in,
                              void* d_out, int out_size, void* d_ws, size_t ws_size,
                              hipStream_t stream) {
  const float* theta = (const float*)d_in[0];
  const float* A     = (const float*)d_in[1];
  float* out = (float*)d_out;

  const int B = in_sizes[0] / (NDIM * NDIM);   // 128
  const size_t lds_bytes =
      (size_t)(2 * NTILE * ROWSTRIDE + NWAVE * 2 * 2 * TILE * TILE) * sizeof(float);

  nw_wavefront_kernel<<<dim3(B), dim3(NWAVE * 32), lds_bytes, stream>>>(theta, A, out);
}